// ContrastModule_44607530336475
// MI455X (gfx1250) — compile-verified
//
#include <hip/hip_runtime.h>
#include <math.h>

// Problem constants (match reference)
#define BB 64
#define PP 1024
#define LL 16
#define HH 256
#define KK 64
#define FLT_BIG 3.402823466e+38f

typedef float v2f __attribute__((ext_vector_type(2)));
typedef float v8f __attribute__((ext_vector_type(8)));

// ---------------------------------------------------------------------------
// Curriculum / temperature helpers (epoch is a device-side scalar input)
// ---------------------------------------------------------------------------
__device__ __forceinline__ float ramp_p(int epoch) {
  if (epoch <= 0) return 0.0f;
  if (epoch >= 80) return 1.0f;
  return (float)epoch / 80.0f;
}

__device__ __forceinline__ float compute_inv_tau(const float* log_inv_tau,
                                                 const int* epoch_p) {
  float p = ramp_p(epoch_p[0]);
  float min_tau = 0.08f - 0.04f * p;
  float it = expf(log_inv_tau[0]);
  it = fminf(it, 100.0f);
  it = fminf(it, 1.0f / fmaxf(min_tau, 1e-6f));
  return it;
}

// ---------------------------------------------------------------------------
// WMMA helper: one wave computes a 16x16 fp32 tile of  A(16xK) @ B(16xK)^T
// using V_WMMA_F32_16X16X4_F32 (A: 2 VGPRs/lane, B: 2 VGPRs/lane, C/D: 8).
// A-frag: lanes 0-15 -> row=lane, K={k,k+1}; lanes 16-31 -> row=lane-16, K={k+2,k+3}
// B-frag: identical layout with "row" = output column (B given row-major).
// D: VGPR r -> (M=r, N=lane) for lanes 0-15, (M=r+8, N=lane-16) for 16-31.
// ---------------------------------------------------------------------------
__device__ __forceinline__ v8f wmma_abT_16x16(const float* __restrict__ A, int lda,
                                              const float* __restrict__ Bm, int ldb,
                                              int K) {
  const int lane = threadIdx.x & 31;
  const int lm = lane & 15;
  const int half = lane >> 4;
  const float* ar = A + lm * lda + 2 * half;
  const float* br = Bm + lm * ldb + 2 * half;
  v8f acc = {};
#pragma unroll 8
  for (int k = 0; k < K; k += 4) {
    v2f a = *(const v2f*)(ar + k);
    v2f b = *(const v2f*)(br + k);
    acc = __builtin_amdgcn_wmma_f32_16x16x4_f32(
        /*neg_a=*/false, a, /*neg_b=*/false, b,
        /*c_mod=*/(short)0, acc, /*reuse_a=*/false, /*reuse_b=*/false);
  }
  return acc;
}

__device__ __forceinline__ void store_tile_16x16(v8f acc, float* __restrict__ D,
                                                 int ldd) {
  const int lane = threadIdx.x & 31;
  const int lm = lane & 15;
  const int half = lane >> 4;
#pragma unroll
  for (int r = 0; r < 8; ++r) D[(r + 8 * half) * ldd + lm] = acc[r];
}

// ---------------------------------------------------------------------------
// Kernel 1: per-batch top-64 by (s1 - s0)  (monotonic in softmax prob).
// Iterative argmax with first-index tie-break (matches lax.top_k ordering rule;
// downstream loss is permutation-invariant over K anyway).
// ---------------------------------------------------------------------------
__global__ void topk_kernel(const float* __restrict__ obj, int* __restrict__ topk) {
  __shared__ float key[PP];
  __shared__ float rv[256];
  __shared__ int ri[256];
  const int b = blockIdx.x;
  const int t = threadIdx.x;
  for (int i = t; i < PP; i += 256) {
    float s0 = obj[(size_t)(b * PP + i) * 2 + 0];
    float s1 = obj[(size_t)(b * PP + i) * 2 + 1];
    key[i] = s1 - s0;
  }
  __syncthreads();
  for (int sel = 0; sel < KK; ++sel) {
    float bv = -FLT_BIG;
    int bi = PP;
    for (int i = t; i < PP; i += 256) {
      float v = key[i];
      if (v > bv) { bv = v; bi = i; }  // increasing i + strict > -> first index wins
    }
    rv[t] = bv; ri[t] = bi;
    __syncthreads();
    for (int s = 128; s > 0; s >>= 1) {
      if (t < s) {
        if (rv[t + s] > rv[t] || (rv[t + s] == rv[t] && ri[t + s] < ri[t])) {
          rv[t] = rv[t + s]; ri[t] = ri[t + s];
        }
      }
      __syncthreads();
    }
    if (t == 0) { topk[b * KK + sel] = ri[0]; key[ri[0]] = -FLT_BIG; }
    __syncthreads();
  }
}

// ---------------------------------------------------------------------------
// Kernel 2: gather top-k feature rows -> contiguous (B*K, H)
// ---------------------------------------------------------------------------
__global__ void gather_kernel(const float* __restrict__ feat,
                              const int* __restrict__ topk,
                              float* __restrict__ featg) {
  const int r = blockIdx.x;            // 0..4095
  const int b = r >> 6;
  const int idx = topk[r];
  const float4* src = (const float4*)(feat + (size_t)(b * PP + idx) * HH);
  float4* dst = (float4*)(featg + (size_t)r * HH);
  dst[threadIdx.x] = src[threadIdx.x]; // 64 threads x float4 = 256 floats
}

// ---------------------------------------------------------------------------
// Kernel 3: Out(R x 256) = In(R x 256) @ W(256 x 256)^T   (wave per tile)
// ---------------------------------------------------------------------------
__global__ void proj_kernel(const float* __restrict__ In,
                            const float* __restrict__ W,
                            float* __restrict__ Out) {
  const int mt = blockIdx.x, nt = blockIdx.y;
  v8f acc = wmma_abT_16x16(In + (size_t)mt * 16 * HH, HH,
                           W + (size_t)nt * 16 * HH, HH, HH);
  store_tile_16x16(acc, Out + (size_t)mt * 16 * HH + nt * 16, HH);
}

// ---------------------------------------------------------------------------
// Kernel 4: row L2-normalize (wave per row, 8 rows per 256-thread block)
// ---------------------------------------------------------------------------
__global__ void normalize_kernel(float* __restrict__ X, int rows) {
  const int w = threadIdx.x >> 5, lane = threadIdx.x & 31;
  const int row = blockIdx.x * 8 + w;
  if (row >= rows) return;
  float* x = X + (size_t)row * HH;
  float v[8];
  float ss = 0.f;
#pragma unroll
  for (int j = 0; j < 8; ++j) { v[j] = x[lane + 32 * j]; ss += v[j] * v[j]; }
  for (int o = 16; o > 0; o >>= 1) ss += __shfl_xor(ss, o, 32);
  float inv = 1.0f / fmaxf(sqrtf(ss), 1e-12f);
#pragma unroll
  for (int j = 0; j < 8; ++j) x[lane + 32 * j] = v[j] * inv;
}

// ---------------------------------------------------------------------------
// Kernel 5: sim[b,l,k] = text[b,l,:] . boxf[b,k,:]   grid (B, K/16), wave/tile
// ---------------------------------------------------------------------------
__global__ void sim_kernel(const float* __restrict__ text,
                           const float* __restrict__ boxf,
                           float* __restrict__ sim) {
  const int b = blockIdx.x, nt = blockIdx.y;
  v8f acc = wmma_abT_16x16(text + (size_t)b * LL * HH, HH,
                           boxf + (size_t)(b * KK + nt * 16) * HH, HH, HH);
  store_tile_16x16(acc, sim + (size_t)b * LL * KK + nt * 16, KK);
}

// Kernel 6: sim_iou[b,k,m] = boxf_iou[b,k,:] . boxf_iou[b,m,:]  grid (B,4,4)
__global__ void sim_iou_kernel(const float* __restrict__ bf,
                               float* __restrict__ sim_iou) {
  const int b = blockIdx.x, kt = blockIdx.y, mt = blockIdx.z;
  v8f acc = wmma_abT_16x16(bf + (size_t)(b * KK + kt * 16) * HH, HH,
                           bf + (size_t)(b * KK + mt * 16) * HH, HH, HH);
  store_tile_16x16(acc, sim_iou + (size_t)b * KK * KK + kt * 16 * KK + mt * 16, KK);
}

// ---------------------------------------------------------------------------
// Kernel 7: per-(b,k) logsumexp of inv_tau * sim_iou row (64 elems), wave/row
// ---------------------------------------------------------------------------
__global__ void lse_iou_kernel(const float* __restrict__ sim_iou,
                               const float* __restrict__ log_inv_tau,
                               const int* __restrict__ epoch,
                               float* __restrict__ lse) {
  const int w = threadIdx.x >> 5, lane = threadIdx.x & 31;
  const int row = blockIdx.x * 8 + w;  // 0..4095
  const float inv = compute_inv_tau(log_inv_tau, epoch);
  const float* r = sim_iou + (size_t)row * KK;
  float t0 = inv * r[lane], t1 = inv * r[lane + 32];
  float mx = fmaxf(t0, t1);
  for (int o = 16; o > 0; o >>= 1) mx = fmaxf(mx, __shfl_xor(mx, o, 32));
  float se = __expf(t0 - mx) + __expf(t1 - mx);
  for (int o = 16; o > 0; o >>= 1) se += __shfl_xor(se, o, 32);
  if (lane == 0) lse[row] = mx + logf(se);
}

// ---------------------------------------------------------------------------
// Kernel 8: per-(b,l) loss terms. Block = 64 threads (thread k per proposal).
// Serial 64-element scalar epilogues are replicated per thread (LDS broadcast).
// ---------------------------------------------------------------------------
__global__ void loss_kernel(const float* __restrict__ pred_center,
                            const float* __restrict__ pred_size,
                            const int* __restrict__ lang_num,
                            const float* __restrict__ ref_center,
                            const int* __restrict__ ref_size_class,
                            const float* __restrict__ ref_size_residual,
                            const float* __restrict__ mean_size_arr,
                            const float* __restrict__ log_inv_tau,
                            const int* __restrict__ epoch_p,
                            const int* __restrict__ topk,
                            const float* __restrict__ sim,
                            const float* __restrict__ sim_iou,
                            const float* __restrict__ lse_iou,
                            float* __restrict__ lang_part,
                            float* __restrict__ iou_part) {
  __shared__ float s_iou[KK], s_w[KK], s_sim[KK], s_term[KK];
  const int bl = blockIdx.x;
  const int b = bl >> 4, l = bl & 15;
  const int k = threadIdx.x;

  if (l >= lang_num[b]) {
    if (k == 0) { lang_part[bl] = 0.f; iou_part[bl] = 0.f; }
    return;
  }

  const float p = ramp_p(epoch_p[0]);
  const float thr = 0.05f + 0.3f * p;
  const float gamma = 1.0f + 2.0f * p;
  const float inv = compute_inv_tau(log_inv_tau, epoch_p);

  // GT box
  const int sc = ref_size_class[b * LL + l];
  float gc[3], gts[3], volg = 1.f;
#pragma unroll
  for (int j = 0; j < 3; ++j) {
    gts[j] = mean_size_arr[sc * 3 + j] + ref_size_residual[(b * LL + l) * 3 + j] + 0.01f;
    gc[j] = ref_center[(b * LL + l) * 3 + j];
    volg *= gts[j];
  }

  // IoU for proposal k
  const int idx = topk[b * KK + k];
  float inter = 1.f, volp = 1.f;
#pragma unroll
  for (int j = 0; j < 3; ++j) {
    float pc = pred_center[(size_t)(b * PP + idx) * 3 + j];
    float ps = pred_size[(size_t)(b * PP + idx) * 3 + j];
    float lo = fmaxf(gc[j] - 0.5f * gts[j], pc - 0.5f * ps);
    float hi = fminf(gc[j] + 0.5f * gts[j], pc + 0.5f * ps);
    inter *= fmaxf(hi - lo, 0.f);
    volp *= ps;
  }
  const float iou = inter / (volg + volp - inter + 1e-7f);
  const float wk = powf(fmaxf(iou, 0.f), gamma) * (iou >= thr ? 1.f : 0.f);
  s_iou[k] = iou;
  s_w[k] = wk;
  s_sim[k] = sim[(size_t)(b * LL + l) * KK + k];
  __syncthreads();

  // Scalar reductions replicated in every thread (LDS broadcasts)
  float S = 0.f;
  for (int m = 0; m < KK; ++m) S += s_w[m];
  int am = 0; float bv = s_iou[0];
  for (int m = 1; m < KK; ++m) if (s_iou[m] > bv) { bv = s_iou[m]; am = m; }
  const bool fb = (S < 1e-6f);
  const float Seff = fb ? 1.f : S;
  const float Ts = Seff / (Seff + 1e-6f);

  // log-softmax over inv*sim
  float tmax = -FLT_BIG;
  for (int m = 0; m < KK; ++m) tmax = fmaxf(tmax, inv * s_sim[m]);
  float se = 0.f;
  for (int m = 0; m < KK; ++m) se += __expf(inv * s_sim[m] - tmax);
  const float lse = tmax + logf(se);

  // ce, s_pos, s_neg
  float ce = 0.f, s_pos = 0.f, s_neg = -FLT_BIG;
  bool anyneg = false;
  for (int m = 0; m < KK; ++m) {
    float wm = fb ? (m == am ? 1.f : 0.f) : s_w[m];
    float tgt = wm / (Seff + 1e-6f);
    float t2 = tgt / (Ts + 1e-6f);
    float logp = inv * s_sim[m] - lse;
    ce -= t2 * logp;
    s_pos += tgt * s_sim[m];
    if (s_iou[m] < 0.05f) { anyneg = true; s_neg = fmaxf(s_neg, s_sim[m]); }
  }
  const float rank = anyneg ? fmaxf(0.2f - s_pos + s_neg, 0.f) : 0.f;
  if (k == 0) lang_part[bl] = ce + 0.5f * rank;

  // iou-aware part: tm = (w w^T) doubly row-renormalized ->
  // tm[k,m] = w_k*w_m*c1_k*c2_k with a=w_k*S, c1=1/(a+eps), c2=1/(a*c1+eps)
  float dot = 0.f;
  const float* simr = sim_iou + (size_t)b * KK * KK + (size_t)k * KK;
  for (int m = 0; m < KK; ++m) dot += s_w[m] * simr[m];
  const float a = wk * S;
  const float c1 = 1.f / (a + 1e-6f);
  const float c2 = 1.f / (a * c1 + 1e-6f);
  const float coef = wk * c1 * c2;
  s_term[k] = coef * (inv * dot - lse_iou[b * KK + k] * S);
  __syncthreads();
  if (k == 0) {
    float t = 0.f;
    for (int m = 0; m < KK; ++m) t += s_term[m];
    iou_part[bl] = -t / (float)KK;
  }
}

// ---------------------------------------------------------------------------
// Kernel 9: fixed-order final reduction (deterministic; no atomics)
// ---------------------------------------------------------------------------
__global__ void reduce_kernel(const float* __restrict__ lang_part,
                              const float* __restrict__ iou_part,
                              float* __restrict__ out) {
  if (threadIdx.x == 0) {
    float a = 0.f, c = 0.f;
    for (int i = 0; i < BB * LL; ++i) { a += lang_part[i]; c += iou_part[i]; }
    out[0] = a / (float)BB;
    out[1] = c / (float)BB;
  }
}

// ---------------------------------------------------------------------------
extern "C" void kernel_launch(void* const* d_in, const int* in_sizes, int n_in,
                              void* d_out, int out_size, void* d_ws, size_t ws_size,
                              hipStream_t stream) {
  (void)in_sizes; (void)n_in; (void)out_size; (void)ws_size;
  const float* pred_center = (const float*)d_in[0];
  const float* pred_size   = (const float*)d_in[1];
  const float* bbox_feat   = (const float*)d_in[2];
  const float* obj_scores  = (const float*)d_in[3];
  const float* lang_emb    = (const float*)d_in[4];
  const int*   lang_num    = (const int*)d_in[5];
  const float* ref_center  = (const float*)d_in[6];
  const int*   ref_szcls   = (const int*)d_in[7];
  const float* ref_szres   = (const float*)d_in[8];
  const float* mean_size   = (const float*)d_in[9];
  const float* W_text      = (const float*)d_in[10];
  const float* W_pc        = (const float*)d_in[11];
  const float* W_pc_iou    = (const float*)d_in[12];
  const float* log_inv_tau = (const float*)d_in[13];
  const int*   epoch       = (const int*)d_in[14];
  float* out = (float*)d_out;

  // Workspace layout (bytes)
  char* ws = (char*)d_ws;
  size_t off = 0;
  int*   topk     = (int*)(ws + off);   off += (size_t)BB * KK * sizeof(int);      // 16 KB
  off = (off + 255) & ~(size_t)255;
  float* featg    = (float*)(ws + off); off += (size_t)BB * KK * HH * sizeof(float); // 4 MB
  float* text     = (float*)(ws + off); off += (size_t)BB * LL * HH * sizeof(float); // 1 MB
  float* boxf     = (float*)(ws + off); off += (size_t)BB * KK * HH * sizeof(float); // 4 MB
  float* boxf_iou = (float*)(ws + off); off += (size_t)BB * KK * HH * sizeof(float); // 4 MB
  float* sim      = (float*)(ws + off); off += (size_t)BB * LL * KK * sizeof(float); // 256 KB
  float* sim_iou  = (float*)(ws + off); off += (size_t)BB * KK * KK * sizeof(float); // 1 MB
  float* lse_iou  = (float*)(ws + off); off += (size_t)BB * KK * sizeof(float);      // 16 KB
  float* lang_part= (float*)(ws + off); off += (size_t)BB * LL * sizeof(float);
  float* iou_part = (float*)(ws + off); off += (size_t)BB * LL * sizeof(float);

  // 1. top-k per batch
  topk_kernel<<<BB, 256, 0, stream>>>(obj_scores, topk);
  // 2. gather features
  gather_kernel<<<BB * KK, 64, 0, stream>>>(bbox_feat, topk, featg);
  // 3. projections (WMMA fp32)
  proj_kernel<<<dim3((BB * LL) / 16, HH / 16), 32, 0, stream>>>(lang_emb, W_text, text);
  proj_kernel<<<dim3((BB * KK) / 16, HH / 16), 32, 0, stream>>>(featg, W_pc, boxf);
  proj_kernel<<<dim3((BB * KK) / 16, HH / 16), 32, 0, stream>>>(featg, W_pc_iou, boxf_iou);
  // 4. row normalize
  normalize_kernel<<<(BB * LL) / 8, 256, 0, stream>>>(text, BB * LL);
  normalize_kernel<<<(BB * KK) / 8, 256, 0, stream>>>(boxf, BB * KK);
  normalize_kernel<<<(BB * KK) / 8, 256, 0, stream>>>(boxf_iou, BB * KK);
  // 5-6. similarity matrices (WMMA fp32)
  sim_kernel<<<dim3(BB, KK / 16), 32, 0, stream>>>(text, boxf, sim);
  sim_iou_kernel<<<dim3(BB, KK / 16, KK / 16), 32, 0, stream>>>(boxf_iou, sim_iou);
  // 7. per-row logsumexp of sim_iou
  lse_iou_kernel<<<(BB * KK) / 8, 256, 0, stream>>>(sim_iou, log_inv_tau, epoch, lse_iou);
  // 8. per-(b,l) losses
  loss_kernel<<<BB * LL, KK, 0, stream>>>(pred_center, pred_size, lang_num,
                                          ref_center, ref_szcls, ref_szres,
                                          mean_size, log_inv_tau, epoch, topk,
                                          sim, sim_iou, lse_iou, lang_part, iou_part);
  // 9. deterministic final reduction
  reduce_kernel<<<1, 32, 0, stream>>>(lang_part, iou_part, out);
}